// diffusion_gcn_90872918049150
// MI455X (gfx1250) — compile-verified
//
#include <hip/hip_runtime.h>

// CDNA5 / gfx1250: wave32, WMMA fp32 path V_WMMA_F32_16X16X4_F32.
typedef float v2f __attribute__((ext_vector_type(2)));
typedef float v8f __attribute__((ext_vector_type(8)));

#define N_NODES 207
#define N_EDGES 1722
#define L_DIM   64

#define ST     210           // LDS row stride in floats (even for b64 loads; 210%64=18 -> low conflicts)
#define M_PAD  224           // 14 M-tiles of 16
#define K_PAD  208           // 52 K-steps of 4 (col 207 zero-padded)

// LDS layout (floats)
#define A_OFF    0
#define XT_OFF   (M_PAD * ST)                 // x transposed: xT[l][v]
#define DEG_OFF  (XT_OFF + L_DIM * ST)
#define COEF_OFF (DEG_OFF + 224)
#define BIAS_OFF (COEF_OFF + 224)
#define LDS_FLOATS (BIAS_OFF + 224)           // 61152 floats = 244608 bytes (< 320 KB/WGP)

static_assert(LDS_FLOATS % 4 == 0, "vector zero-fill");
static_assert(LDS_FLOATS * 4 <= 320 * 1024, "fits CDNA5 LDS");

__global__ void __launch_bounds__(256, 1)
diffusion_gcn_kernel(const float* __restrict__ inputs,          // (1024, 2, 207, 64)
                     const float* __restrict__ weight_diff,      // (288, 1722)
                     const float* __restrict__ bias_diffusion,   // (288, 207)
                     const float* __restrict__ weight_self_loop, // (288, 207)
                     const int*   __restrict__ ind,              // (1024,)
                     const int*   __restrict__ edge_i,           // (1722,)
                     const int*   __restrict__ edge_j,           // (1722,)
                     float*       __restrict__ out)              // (1024, 207, 64)
{
    extern __shared__ float lds[];
    const int b   = blockIdx.x;
    const int tid = threadIdx.x;
    const int indb = ind[b];

    // ---- Phase 0: zero LDS (A, xT, deg, coef, bias) ----
    {
        float4* l4 = reinterpret_cast<float4*>(lds);
        for (int i = tid; i < LDS_FLOATS / 4; i += 256)
            l4[i] = make_float4(0.f, 0.f, 0.f, 0.f);
    }
    __syncthreads();

    // ---- Phase 1a: scatter edges into dense A (symmetric) + deg, via LDS atomics ----
    {
        const float* wrow = weight_diff + (size_t)indb * N_EDGES;
        for (int e = tid; e < N_EDGES; e += 256) {
            const int i = edge_i[e];
            const int j = edge_j[e];
            const float w = wrow[e];
            atomicAdd(&lds[A_OFF + i * ST + j], w);
            atomicAdd(&lds[A_OFF + j * ST + i], w);
            atomicAdd(&lds[DEG_OFF + i], w);
            atomicAdd(&lds[DEG_OFF + j], w);
        }
    }

    // ---- Phase 1b: stage x (channel 0) transposed into LDS: xT[l][v] ----
    {
        const float* xb = inputs + (size_t)b * (2 * N_NODES * L_DIM);
        for (int idx = tid; idx < N_NODES * L_DIM; idx += 256) {
            const int v = idx >> 6;       // node
            const int l = idx & 63;       // feature
            lds[XT_OFF + l * ST + v] = xb[idx];
        }
    }
    __syncthreads();

    // ---- Phase 2: per-node epilogue coefficients ----
    for (int w = tid; w < N_NODES; w += 256) {
        lds[COEF_OFF + w] = 1.0f + lds[DEG_OFF + w]
                          + weight_self_loop[(size_t)indb * N_NODES + w];
        lds[BIAS_OFF + w] = bias_diffusion[(size_t)indb * N_NODES + w];
    }
    __syncthreads();

    // ---- Phase 3: acc = A @ x via V_WMMA_F32_16X16X4_F32, fused epilogue ----
    const int lane = tid & 31;       // wave32
    const int wv   = tid >> 5;       // 8 waves
    const int half = lane >> 4;      // lane group (0: lanes 0-15, 1: lanes 16-31)
    const int lm   = lane & 15;

    const float* Amat = lds + A_OFF;
    const float* XT   = lds + XT_OFF;
    float* ob = out + (size_t)b * (N_NODES * L_DIM);

    // 14 M-tiles x 4 N-tiles = 56 tiles, 7 per wave. Wave-uniform loop: EXEC all-ones for WMMA.
    for (int t = wv; t < 56; t += 8) {
        const int mt = t >> 2;
        const int nt = t & 3;
        const float* arow = Amat + (mt * 16 + lm) * ST;   // A-frag: M = lane%16
        const float* brow = XT   + (nt * 16 + lm) * ST;   // B-frag: N = lane%16 (xT rows)

        v8f acc = {0.f, 0.f, 0.f, 0.f, 0.f, 0.f, 0.f, 0.f};
        for (int kb = 0; kb < K_PAD; kb += 4) {
            // ISA 7.12.2: 32-bit A 16x4 -> lane (k>>1)*16+m holds k = 2*half+{0,1}; B mirrored.
            const v2f afrag = *reinterpret_cast<const v2f*>(arow + kb + 2 * half);
            const v2f bfrag = *reinterpret_cast<const v2f*>(brow + kb + 2 * half);
            acc = __builtin_amdgcn_wmma_f32_16x16x4_f32(
                false, afrag, false, bfrag, (short)0, acc, false, false);
        }

        // C/D layout: VGPR r holds M = r + 8*half, N = lane%16 (within tile)
        const int l = nt * 16 + lm;
        #pragma unroll
        for (int r = 0; r < 8; ++r) {
            const int w = mt * 16 + half * 8 + r;
            if (w < N_NODES) {
                const float xv = XT[l * ST + w];
                ob[w * L_DIM + l] = lds[COEF_OFF + w] * xv - acc[r] + lds[BIAS_OFF + w];
            }
        }
    }
}

extern "C" void kernel_launch(void* const* d_in, const int* in_sizes, int n_in,
                              void* d_out, int out_size, void* d_ws, size_t ws_size,
                              hipStream_t stream) {
    (void)in_sizes; (void)n_in; (void)out_size; (void)d_ws; (void)ws_size;

    const float* inputs           = (const float*)d_in[0];
    const float* weight_diff      = (const float*)d_in[1];
    const float* bias_diffusion   = (const float*)d_in[2];
    const float* weight_self_loop = (const float*)d_in[3];
    const int*   ind              = (const int*)d_in[4];
    const int*   edge_i           = (const int*)d_in[5];
    const int*   edge_j           = (const int*)d_in[6];
    float* out = (float*)d_out;

    const size_t ldsBytes = (size_t)LDS_FLOATS * sizeof(float);   // ~244 KB
    // Not a stream op; safe (and deterministic) under graph capture.
    hipFuncSetAttribute((const void*)diffusion_gcn_kernel,
                        hipFuncAttributeMaxDynamicSharedMemorySize, (int)ldsBytes);

    diffusion_gcn_kernel<<<1024, 256, ldsBytes, stream>>>(
        inputs, weight_diff, bias_diffusion, weight_self_loop,
        ind, edge_i, edge_j, out);
}